// STFT_18648747999406
// MI455X (gfx1250) — compile-verified
//
#include <hip/hip_runtime.h>
#include <math.h>
#include <stdint.h>

typedef __attribute__((ext_vector_type(2))) float v2f;
typedef __attribute__((ext_vector_type(4))) float v4f;
typedef __attribute__((ext_vector_type(8))) float v8f;
typedef __attribute__((ext_vector_type(4))) unsigned int u32x4;
typedef __attribute__((ext_vector_type(4))) int i32x4;
typedef __attribute__((ext_vector_type(8))) int i32x8;

#define FL        800
#define HOP       200
#define NCH       802
#define NFRAMES   801
#define FPAD      4            // front pad in the frame dimension of ft
#define FSTRIDE   820          // FPAD + 816 : frame f lives at index f+FPAD
#define NSAMP     160000
#define NBATCH    32
#define PADW      400
#define NPAD      160800
#define TINYF     1.1754944e-38f
#define WINSPAN   (15 * HOP + FL)   // 3800 floats: 16 overlapping windows

// ---------------------------------------------------------------------------
// Stage 0a: zero the 4-float front pad of every (b,c) row of ft
// (tail pad [805,820) is written as zeros by the last forward frame-tile)
// ---------------------------------------------------------------------------
__global__ void __launch_bounds__(256) padzero_kernel(float* __restrict__ ft) {
  int i = blockIdx.x * 256 + threadIdx.x;          // one thread per (b,c) row
  if (i < NBATCH * NCH) {
    v4f z = {};
    *(v4f*)(ft + (size_t)i * FSTRIDE) = z;         // 16B-aligned (820*4 = 205*16)
  }
}

// ---------------------------------------------------------------------------
// Stage 0b: window_sumsquare for hann(800), hop 200, 801 frames -> wsum[160800]
// ---------------------------------------------------------------------------
__global__ void __launch_bounds__(256) wsum_kernel(float* __restrict__ wsum) {
  int t = blockIdx.x * 256 + threadIdx.x;
  if (t >= NPAD) return;
  int g = t / HOP;
  float w = 0.0f;
  #pragma unroll
  for (int d = 0; d < 4; ++d) {
    int f = g - d;
    if (f >= 0 && f < NFRAMES) {
      int tau = t - HOP * f;                       // always in [0, 800)
      float h = 0.5f - 0.5f * cosf(6.2831853071795864769f *
                                   (float)tau * (1.0f / (float)FL));
      w += h * h;
    }
  }
  wsum[t] = w;
}

// ---------------------------------------------------------------------------
// Stage 1: forward STFT as implicit GEMM with V_WMMA_F32_16X16X4_F32
//   ft[b][c][f+FPAD] = sum_k window[f][k] * fb[c][k]
//   grid = (7 chan-supertiles, 51 frame-tiles, 32 batches), 256 threads
//   Interior frame-tiles stage the 3800-float window via the Tensor Data
//   Mover (tensor_load_to_lds + s_wait_tensorcnt); edge tiles reflect-fill.
// ---------------------------------------------------------------------------
__global__ void __launch_bounds__(256) stft_fwd(const float* __restrict__ x,
                                                const float* __restrict__ fb,
                                                float* __restrict__ ft) {
  __shared__ float awin[WINSPAN];
  const int b    = blockIdx.z;
  const int f0   = blockIdx.y * 16;
  const int base = f0 * HOP;                       // start in padded domain

#if __has_builtin(__builtin_amdgcn_tensor_load_to_lds)
  const bool interior = (blockIdx.y >= 1) && (blockIdx.y <= 48);
  if (interior) {
    // whole window is in-range: straight 15200B global->LDS DMA via TDM
    if (threadIdx.x < 32) {                        // one wave issues the DMA
      const uint64_t ga  = (uint64_t)(uintptr_t)(x + (size_t)b * NSAMP +
                                                 (base - PADW));
      const uint32_t lds = (uint32_t)(uintptr_t)(void*)awin;  // LDS offset
      u32x4 g0;
      g0[0] = 1u;                                            // count=1, user D#
      g0[1] = lds;                                           // lds_addr
      g0[2] = (uint32_t)(ga & 0xFFFFFFFFull);                // global_addr lo
      g0[3] = (uint32_t)((ga >> 32) & 0x01FFFFFFull)         // global_addr hi
              | (2u << 30);                                  // type=2 (image)
      i32x8 g1 = {};
      g1[0] = (2 << 16);                           // data_size = 4 bytes
      g1[1] = (int)((WINSPAN & 0xFFFF) << 16);     // tensor_dim0[15:0]
      g1[2] = (int)((WINSPAN >> 16) | (1 << 16));  // tensor_dim0[31:16], dim1=1
      g1[3] = (int)(WINSPAN << 16);                // tile_dim0 = 3800
      g1[4] = 1;                                   // tile_dim1 = 1
      g1[5] = WINSPAN;                             // tensor_dim0_stride lo32
      g1[6] = (int)((WINSPAN & 0xFFFF) << 16);     // dim0_str hi=0 | dim1_str lo
      g1[7] = 0;
      i32x4 gz = {};
#if defined(__clang_major__) && (__clang_major__ >= 23)
      i32x8 gz8 = {};
      __builtin_amdgcn_tensor_load_to_lds(g0, g1, gz, gz, gz8, 0);
#else
      __builtin_amdgcn_tensor_load_to_lds(g0, g1, gz, gz, 0);
#endif
      __builtin_amdgcn_s_wait_tensorcnt(0);
    }
  } else
#endif
  {
    for (int s = threadIdx.x; s < WINSPAN; s += 256) {
      int j = base + s - PADW;
      j = (j < 0) ? -j : j;
      j = (j >= NSAMP) ? (2 * NSAMP - 2 - j) : j;
      awin[s] = x[(size_t)b * NSAMP + j];
    }
  }
  __syncthreads();

  const int wave  = threadIdx.x >> 5;
  const int lane  = threadIdx.x & 31;
  const int ctile = blockIdx.x * 8 + wave;
  if (ctile >= 51) return;                         // whole wave drops out

  const int half = lane >> 4;                      // 0: K+0/K+1, 1: K+2/K+3
  const int koff = half << 1;
  const int rown = lane & 15;                      // A: row M; B: col N

  const int cg = ctile * 16 + rown;                // channel for B fragment
  const int cl = (cg > NCH - 1) ? (NCH - 1) : cg;  // clamp loads, guard stores
  const float* bp = fb + (size_t)cl * FL + koff;
  const float* ap = awin + rown * HOP + koff;

  v8f acc = {};
  #pragma unroll 4
  for (int k = 0; k < FL; k += 4) {
    v2f a  = *(const v2f*)(ap + k);                // A[M=row][K=k+koff, +1]
    v2f bv = *(const v2f*)(bp + k);                // B[K=k+koff, +1][N=chan]
    acc = __builtin_amdgcn_wmma_f32_16x16x4_f32(
        false, a, false, bv, (short)0, acc, false, false);
  }

  if (cg < NCH) {
    const int fb0 = f0 + (half << 3);              // C rows: v -> M = v / v+8
    if (fb0 + 8 > NFRAMES) {                       // only last frame-tile
      #pragma unroll
      for (int v = 0; v < 8; ++v)
        if (fb0 + v >= NFRAMES) acc[v] = 0.0f;     // also zero-fills tail pad
    }
    float* op = ft + (size_t)b * NCH * FSTRIDE + (size_t)cg * FSTRIDE +
                (fb0 + FPAD);                      // 16B aligned
    v4f lo = {acc[0], acc[1], acc[2], acc[3]};
    v4f hi = {acc[4], acc[5], acc[6], acc[7]};
    *(v4f*)op       = lo;
    *(v4f*)(op + 4) = hi;
  }
}

// ---------------------------------------------------------------------------
// Stage 2: inverse STFT as GEMM with V_WMMA_F32_16X16X4_F32
//   y[(b,g), r] = sum_c sum_{d=0..3} ft[b,c,g-d] * ib[c][r + 200 d]
//   (frame pads in ft are zero -> unconditional A loads)
//   then /= wsum (where > TINY), *4, slice [400:-400]
//   grid = (13 r-tiles, 200 mtile-groups), 8 waves/block, one 16x16 tile/wave
// ---------------------------------------------------------------------------
__global__ void __launch_bounds__(256) istft_inv(const float* __restrict__ ft,
                                                 const float* __restrict__ ib,
                                                 const float* __restrict__ wsum,
                                                 float* __restrict__ out) {
  const int wave  = threadIdx.x >> 5;
  const int lane  = threadIdx.x & 31;
  const int ntile = blockIdx.x;                    // 0..12 (r tiles, 200->208)
  const int mtile = blockIdx.y * 8 + wave;         // 0..1599
  const int half  = lane >> 4;
  const int koff  = half << 1;                     // d = koff, koff+1
  const int rown  = lane & 15;

  // 800 % 16 == 0, so each tile sits inside one batch: b, g-base are uniform
  const int m0  = mtile * 16;
  const int b   = m0 / 800;
  const int g   = (m0 % 800) + 2 + rown;           // g in [2, 801]
  const int idx0 = g - koff + FPAD;                // frame d=koff (padded idx)
  const float* ftb = ft + (size_t)b * NCH * FSTRIDE;

  const int r   = ntile * 16 + rown;               // offset within 200-block
  const int rcl = (r < HOP) ? r : (HOP - 1);
  const int t0  = rcl + HOP * koff;                // tau for d = koff

  v8f acc = {};
  #pragma unroll 4
  for (int c = 0; c < NCH; ++c) {
    const float* fp = ftb + (size_t)c * FSTRIDE;
    v2f a;
    a[0] = fp[idx0];                               // ft[b,c,g-koff]
    a[1] = fp[idx0 - 1];                           // ft[b,c,g-koff-1]
    const float* ip = ib + (size_t)c * FL;
    v2f bv;
    bv[0] = ip[t0];                                // ib[c][r + 200*koff]
    bv[1] = ip[t0 + HOP];                          // ib[c][r + 200*(koff+1)]
    acc = __builtin_amdgcn_wmma_f32_16x16x4_f32(
        false, a, false, bv, (short)0, acc, false, false);
  }

  if (r < HOP) {
    const int gg0 = (m0 % 800) + 2 + (half << 3);  // uniform per half-wave
    const int tb  = HOP * gg0 + r;                 // t in [400, 160400)
    const float* wp = wsum + tb;
    float* op = out + (size_t)b * NSAMP + (tb - PADW);
    #pragma unroll
    for (int v = 0; v < 8; ++v) {                  // stride-200 imm offsets
      float w   = wp[v * HOP];
      float val = acc[v];
      if (w > TINYF) val /= w;
      op[v * HOP] = val * 4.0f;
    }
  }
}

// ---------------------------------------------------------------------------
extern "C" void kernel_launch(void* const* d_in, const int* in_sizes, int n_in,
                              void* d_out, int out_size, void* d_ws, size_t ws_size,
                              hipStream_t stream) {
  const float* x  = (const float*)d_in[0];   // (32, 160000)
  const float* fb = (const float*)d_in[1];   // (802, 1, 800)
  const float* ib = (const float*)d_in[2];   // (802, 1, 800)
  float* out = (float*)d_out;                // (32, 1, 160000) flat

  float* ft   = (float*)d_ws;                              // 32*802*820 floats
  float* wsum = ft + (size_t)NBATCH * NCH * FSTRIDE;       // 160800 floats

  padzero_kernel<<<dim3((NBATCH * NCH + 255) / 256), dim3(256), 0, stream>>>(ft);
  wsum_kernel   <<<dim3((NPAD + 255) / 256),         dim3(256), 0, stream>>>(wsum);
  stft_fwd      <<<dim3(7, 51, NBATCH),              dim3(256), 0, stream>>>(x, fb, ft);
  istft_inv     <<<dim3(13, 200),                    dim3(256), 0, stream>>>(ft, ib, wsum, out);
}